// GINConvNet_78658031059345
// MI455X (gfx1250) — compile-verified
//
#include <hip/hip_runtime.h>
#include <hip/hip_bf16.h>
#include <math.h>

// ---------------- WMMA types ----------------
typedef float v2f __attribute__((ext_vector_type(2)));
typedef float v8f __attribute__((ext_vector_type(8)));

#define NN 100000
#define NE 1000000
#define NG 2048
#define FI 78
#define DIM 32
#define TD 208
#define BN_EPS 1e-5f

// D = A(16xK) * B(Kx16) + C using V_WMMA_F32_16X16X4_F32.
// Layouts (wave32):
//  A 16x4 : lane&15 = M row; K = kc + 2*(lane>>4) + vgpr
//  B 4x16 : lane&15 = N col; K = kc + 2*(lane>>4) + vgpr
//  C/D    : lane&15 = N col; M = vgpr + 8*(lane>>4)
__device__ inline v8f wmma4(v2f a, v2f b, v8f c) {
    return __builtin_amdgcn_wmma_f32_16x16x4_f32(false, a, false, b, (short)0, c,
                                                 false, false);
}

// ---------------- small utility kernels ----------------
__global__ void k_fill(float* __restrict__ p, float v, long n) {
    long i = (long)blockIdx.x * blockDim.x + threadIdx.x;
    if (i < n) p[i] = v;
}

__global__ void k_copy(float* __restrict__ dst, const float* __restrict__ src, long n) {
    long i = (long)blockIdx.x * blockDim.x + threadIdx.x;
    if (i < n) dst[i] = src[i];
}

// zero-padded 2D copy: dst[rdst x cdst] <- src[rsrc x csrc]
__global__ void k_pad2d(float* __restrict__ dst, const float* __restrict__ src,
                        int rdst, int cdst, int rsrc, int csrc) {
    int i = blockIdx.x * blockDim.x + threadIdx.x;
    if (i >= rdst * cdst) return;
    int r = i / cdst, c = i % cdst;
    dst[i] = (r < rsrc && c < csrc) ? src[r * csrc + c] : 0.0f;
}

// agg78[n,0:80] = {x[n,0:78], 0, 0}
__global__ void k_init_agg78(float* __restrict__ agg, const float* __restrict__ x) {
    long i = (long)blockIdx.x * blockDim.x + threadIdx.x;
    if (i >= (long)NN * 80) return;
    int n = (int)(i / 80), f = (int)(i % 80);
    agg[i] = (f < FI) ? x[(long)n * FI + f] : 0.0f;
}

// one wave per edge: agg[dst,:] += feat[src,:]
__global__ void k_scatter(const float* __restrict__ feat, int fdim, int fstride,
                          float* __restrict__ agg, int astride,
                          const int* __restrict__ src, const int* __restrict__ dst,
                          int nE) {
    int wid = (blockIdx.x * blockDim.x + threadIdx.x) >> 5;
    if (wid >= nE) return;
    int lane = threadIdx.x & 31;
    int s = src[wid], d = dst[wid];
    for (int f = lane; f < fdim; f += 32)
        atomicAdd(&agg[(long)d * astride + f], feat[(long)s * fstride + f]);
}

// ---------------- fused GIN node MLP ----------------
// OUT[16 nodes x 32] = act2( relu( AGG(16xK1) @ W1(K1x32) + b1 ) @ W2(32x32) + b2 )
// one wave per 16-node tile; intermediate staged via LDS for layout transpose.
__global__ void k_node_mlp(const float* __restrict__ AGG, int lda, int K1,
                           const float* __restrict__ W1, const float* __restrict__ b1,
                           const float* __restrict__ W2, const float* __restrict__ b2,
                           float* __restrict__ OUT, int nTiles, int act2) {
    __shared__ float lds[8][16 * 32];
    int wid  = (blockIdx.x * blockDim.x + threadIdx.x) >> 5;
    int w    = threadIdx.x >> 5;
    int lane = threadIdx.x & 31;
    int tile = (wid < nTiles) ? wid : (nTiles - 1);  // clamp: keep all waves for barrier
    int m0   = tile * 16;
    int row  = lane & 15;            // A row / C column index
    int col  = lane & 15;
    int kg   = (lane >> 4) << 1;     // K sub-offset for A/B frags
    int mb   = (lane >> 4) << 3;     // M base for C/D frags

    // GEMM1: AGG @ W1 + b1, relu
    v8f acc[2];
    for (int t = 0; t < 2; ++t) {
        float bv = b1[t * 16 + col];
        for (int r = 0; r < 8; ++r) acc[t][r] = bv;
    }
    for (int kc = 0; kc < K1; kc += 4) {
        v2f a;
        a.x = AGG[(long)(m0 + row) * lda + kc + kg];
        a.y = AGG[(long)(m0 + row) * lda + kc + kg + 1];
        for (int t = 0; t < 2; ++t) {
            v2f b;
            b.x = W1[(kc + kg) * 32 + t * 16 + col];
            b.y = W1[(kc + kg + 1) * 32 + t * 16 + col];
            acc[t] = wmma4(a, b, acc[t]);
        }
    }
    for (int t = 0; t < 2; ++t)
        for (int r = 0; r < 8; ++r) {
            float v = acc[t][r];
            lds[w][(mb + r) * 32 + t * 16 + col] = v > 0.0f ? v : 0.0f;
        }
    __syncthreads();

    // GEMM2: lds @ W2 + b2, act2 (0=relu, 1=elu)
    v8f acc2[2];
    for (int t = 0; t < 2; ++t) {
        float bv = b2[t * 16 + col];
        for (int r = 0; r < 8; ++r) acc2[t][r] = bv;
    }
    for (int kc = 0; kc < 32; kc += 4) {
        v2f a;
        a.x = lds[w][row * 32 + kc + kg];
        a.y = lds[w][row * 32 + kc + kg + 1];
        for (int t = 0; t < 2; ++t) {
            v2f b;
            b.x = W2[(kc + kg) * 32 + t * 16 + col];
            b.y = W2[(kc + kg + 1) * 32 + t * 16 + col];
            acc2[t] = wmma4(a, b, acc2[t]);
        }
    }
    if (wid < nTiles) {
        for (int t = 0; t < 2; ++t)
            for (int r = 0; r < 8; ++r) {
                float v = acc2[t][r];
                v = (act2 == 1) ? (v > 0.0f ? v : (expf(v) - 1.0f))
                                : (v > 0.0f ? v : 0.0f);
                OUT[(long)(m0 + mb + r) * 32 + t * 16 + col] = v;
            }
    }
}

// ---------------- generic WMMA GEMM: C = act(A @ B + bias) ----------------
__global__ void k_gemm(const float* __restrict__ A, int lda,
                       const float* __restrict__ B, int ldb,
                       const float* __restrict__ bias,
                       float* __restrict__ C, int ldc,
                       int K, int Mtiles, int Ntiles, int act) {
    int wid = (blockIdx.x * blockDim.x + threadIdx.x) >> 5;
    if (wid >= Mtiles * Ntiles) return;
    int lane = threadIdx.x & 31;
    int mi = wid / Ntiles, ni = wid % Ntiles;
    int m0 = mi * 16, n0 = ni * 16;
    int row = lane & 15, col = lane & 15;
    int kg = (lane >> 4) << 1;
    int mb = (lane >> 4) << 3;

    v8f acc;
    float bv = bias ? bias[n0 + col] : 0.0f;
    for (int r = 0; r < 8; ++r) acc[r] = bv;
    for (int kc = 0; kc < K; kc += 4) {
        v2f a, b;
        a.x = A[(long)(m0 + row) * lda + kc + kg];
        a.y = A[(long)(m0 + row) * lda + kc + kg + 1];
        b.x = B[(long)(kc + kg) * ldb + n0 + col];
        b.y = B[(long)(kc + kg + 1) * ldb + n0 + col];
        acc = wmma4(a, b, acc);
    }
    for (int r = 0; r < 8; ++r) {
        float v = acc[r];
        if (act == 1) v = v > 0.0f ? v : 0.0f;
        C[(long)(m0 + mb + r) * ldc + n0 + col] = v;
    }
}

// ---------------- BatchNorm (training-mode, biased stats) ----------------
// one block per column: S[c] = sum, S[cols+c] = sum of squares
__global__ void k_col_stats(const float* __restrict__ X, float* __restrict__ S,
                            int rows, int cols) {
    __shared__ float s1[256], s2[256];
    int c = blockIdx.x, tid = threadIdx.x;
    float a = 0.0f, b = 0.0f;
    for (int r = tid; r < rows; r += blockDim.x) {
        float v = X[(long)r * cols + c];
        a += v; b += v * v;
    }
    s1[tid] = a; s2[tid] = b;
    __syncthreads();
    for (int off = 128; off > 0; off >>= 1) {
        if (tid < off) { s1[tid] += s1[tid + off]; s2[tid] += s2[tid + off]; }
        __syncthreads();
    }
    if (tid == 0) { S[c] = s1[0]; S[cols + c] = s2[0]; }
}

__global__ void k_bn_apply(const float* __restrict__ X, float* __restrict__ Y,
                           const float* __restrict__ S, int rows, int cols,
                           const float* __restrict__ gamma,
                           const float* __restrict__ beta) {
    long i = (long)blockIdx.x * blockDim.x + threadIdx.x;
    if (i >= (long)rows * cols) return;
    int c = (int)(i % cols);
    float m = S[c] / (float)rows;
    float v = S[cols + c] / (float)rows - m * m;
    float inv = rsqrtf((v > 0.0f ? v : 0.0f) + BN_EPS);
    float y = (X[i] - m) * inv;
    if (gamma) y = y * gamma[c] + beta[c];
    Y[i] = y;
}

// ---------------- graph pooling ----------------
__global__ void k_pool(const float* __restrict__ H, const int* __restrict__ batch,
                       float* __restrict__ G) {
    long i = (long)blockIdx.x * blockDim.x + threadIdx.x;
    if (i >= (long)NN * DIM) return;
    int n = (int)(i >> 5), c = (int)(i & 31);
    atomicAdd(&G[(long)batch[n] * DIM + c], H[i]);
}

// ---------------- softmax over 128 cols, one wave per row ----------------
__global__ void k_softmax128(float* __restrict__ T, int rows) {
    int wid = (blockIdx.x * blockDim.x + threadIdx.x) >> 5;
    if (wid >= rows) return;
    int lane = threadIdx.x & 31;
    float v[4], mx = -3.4e38f;
    for (int j = 0; j < 4; ++j) {
        v[j] = T[(long)wid * 128 + lane + 32 * j];
        mx = fmaxf(mx, v[j]);
    }
    for (int off = 16; off > 0; off >>= 1) mx = fmaxf(mx, __shfl_xor(mx, off, 32));
    float s = 0.0f;
    for (int j = 0; j < 4; ++j) { v[j] = expf(v[j] - mx); s += v[j]; }
    for (int off = 16; off > 0; off >>= 1) s += __shfl_xor(s, off, 32);
    float inv = 1.0f / s;
    for (int j = 0; j < 4; ++j) T[(long)wid * 128 + lane + 32 * j] = v[j] * inv;
}

// xc = [g | t]
__global__ void k_build_xc(float* __restrict__ XC, const float* __restrict__ G,
                           const float* __restrict__ T) {
    long i = (long)blockIdx.x * blockDim.x + threadIdx.x;
    if (i >= (long)NG * 256) return;
    int r = (int)(i >> 8), c = (int)(i & 255);
    XC[i] = (c < 128) ? G[(long)r * 128 + c] : T[(long)r * 128 + (c - 128)];
}

// out[row] = sigmoid(dot(XC2[row,:], W5) + b5), one wave per row
__global__ void k_head(const float* __restrict__ XC2, const float* __restrict__ W5,
                       const float* __restrict__ b5, float* __restrict__ out, int rows) {
    int wid = (blockIdx.x * blockDim.x + threadIdx.x) >> 5;
    if (wid >= rows) return;
    int lane = threadIdx.x & 31;
    float s = 0.0f;
    for (int j = 0; j < 4; ++j) {
        int c = lane + 32 * j;
        s += XC2[(long)wid * 128 + c] * W5[c];
    }
    for (int off = 16; off > 0; off >>= 1) s += __shfl_xor(s, off, 32);
    if (lane == 0) out[wid] = 1.0f / (1.0f + expf(-(s + b5[0])));
}

// ======================================================================
extern "C" void kernel_launch(void* const* d_in, const int* in_sizes, int n_in,
                              void* d_out, int out_size, void* d_ws, size_t ws_size,
                              hipStream_t stream) {
    const float* x      = (const float*)d_in[0];
    const int*   eidx   = (const int*)d_in[1];
    const int*   batch  = (const int*)d_in[2];
    const float* target = (const float*)d_in[3];
    const float* W11a = (const float*)d_in[4];  const float* b11a = (const float*)d_in[5];
    const float* W11b = (const float*)d_in[6];  const float* b11b = (const float*)d_in[7];
    const float* W12a = (const float*)d_in[8];  const float* b12a = (const float*)d_in[9];
    const float* W12b = (const float*)d_in[10]; const float* b12b = (const float*)d_in[11];
    const float* W13a = (const float*)d_in[12]; const float* b13a = (const float*)d_in[13];
    const float* W13b = (const float*)d_in[14]; const float* b13b = (const float*)d_in[15];
    const float* Wn2  = (const float*)d_in[16]; const float* bn2b = (const float*)d_in[17];
    const float* g1   = (const float*)d_in[18]; const float* be1  = (const float*)d_in[19];
    const float* W31  = (const float*)d_in[20]; const float* b31  = (const float*)d_in[21];
    const float* W32  = (const float*)d_in[22]; const float* b32  = (const float*)d_in[23];
    const float* W4   = (const float*)d_in[24]; const float* b4   = (const float*)d_in[25];
    const float* W5   = (const float*)d_in[26]; const float* b5   = (const float*)d_in[27];
    const int* esrc = eidx;
    const int* edst = eidx + NE;
    float* outp = (float*)d_out;   // [0,2048): out ; [2048, 2048+2048*128): g

    // ---- workspace layout ----
    float* ws = (float*)d_ws;
    size_t off = 0;
    auto alloc = [&](size_t n) { float* p = ws + off; off += n; return p; };
    float* A78  = alloc((size_t)NN * 80);
    float* AGG  = alloc((size_t)NN * DIM);
    float* HP   = alloc((size_t)NN * DIM);
    float* HN   = alloc((size_t)NN * DIM);
    float* W1P  = alloc(80 * 32);
    float* S    = alloc(2 * DIM);
    float* TS   = alloc(2 * TD);
    float* WP31 = alloc(208 * 176);
    float* BP31 = alloc(176);
    float* WP32 = alloc(176 * 128);
    float* GRAW = alloc((size_t)NG * DIM);
    float* G    = alloc((size_t)NG * 128);
    float* TN   = alloc((size_t)NG * TD);
    float* T1   = alloc((size_t)NG * 176);
    float* T2   = alloc((size_t)NG * 128);
    float* XC   = alloc((size_t)NG * 256);
    float* XC2  = alloc((size_t)NG * 128);
    (void)ws_size; (void)n_in; (void)in_sizes; (void)out_size;

    const int B = 256;
    auto g1d = [](long n) { return (unsigned)((n + 255) / 256); };

    // ---- padded weight copies ----
    k_pad2d<<<g1d(80 * 32), B, 0, stream>>>(W1P, W11a, 80, 32, 78, 32);
    k_pad2d<<<g1d(208 * 176), B, 0, stream>>>(WP31, W31, 208, 176, 208, 170);
    k_pad2d<<<g1d(176), B, 0, stream>>>(BP31, b31, 1, 176, 1, 170);
    k_pad2d<<<g1d(176 * 128), B, 0, stream>>>(WP32, W32, 176, 128, 170, 128);

    const int nTiles = NN / 16;                // 6250
    const unsigned mlpBlocks = (nTiles + 7) / 8;
    const unsigned edgeBlocks = (NE * 32) / B; // wave per edge

    // ---- GIN layer 1 (78-dim aggregate) ----
    k_init_agg78<<<g1d((long)NN * 80), B, 0, stream>>>(A78, x);
    k_scatter<<<edgeBlocks, B, 0, stream>>>(x, FI, FI, A78, 80, esrc, edst, NE);
    k_node_mlp<<<mlpBlocks, B, 0, stream>>>(A78, 80, 80, W1P, b11a, W11b, b11b, HP, nTiles, 1);
    k_col_stats<<<DIM, B, 0, stream>>>(HP, S, NN, DIM);
    k_bn_apply<<<g1d((long)NN * DIM), B, 0, stream>>>(HP, HN, S, NN, DIM, nullptr, nullptr);

    // ---- GIN layer 2 ----
    k_copy<<<g1d((long)NN * DIM), B, 0, stream>>>(AGG, HN, (long)NN * DIM);
    k_scatter<<<edgeBlocks, B, 0, stream>>>(HN, DIM, DIM, AGG, DIM, esrc, edst, NE);
    k_node_mlp<<<mlpBlocks, B, 0, stream>>>(AGG, DIM, DIM, W12a, b12a, W12b, b12b, HP, nTiles, 0);
    k_col_stats<<<DIM, B, 0, stream>>>(HP, S, NN, DIM);
    k_bn_apply<<<g1d((long)NN * DIM), B, 0, stream>>>(HP, HN, S, NN, DIM, nullptr, nullptr);

    // ---- GIN layer 3 ----
    k_copy<<<g1d((long)NN * DIM), B, 0, stream>>>(AGG, HN, (long)NN * DIM);
    k_scatter<<<edgeBlocks, B, 0, stream>>>(HN, DIM, DIM, AGG, DIM, esrc, edst, NE);
    k_node_mlp<<<mlpBlocks, B, 0, stream>>>(AGG, DIM, DIM, W13a, b13a, W13b, b13b, HP, nTiles, 0);
    k_col_stats<<<DIM, B, 0, stream>>>(HP, S, NN, DIM);
    k_bn_apply<<<g1d((long)NN * DIM), B, 0, stream>>>(HP, HN, S, NN, DIM, nullptr, nullptr);

    // ---- pooling + g = relu(g @ Wn2 + bn2b) ----
    k_fill<<<g1d((long)NG * DIM), B, 0, stream>>>(GRAW, 0.0f, (long)NG * DIM);
    k_pool<<<g1d((long)NN * DIM), B, 0, stream>>>(HN, batch, GRAW);
    k_gemm<<<(128 * 8 + 7) / 8, B, 0, stream>>>(GRAW, DIM, Wn2, 128, bn2b, G, 128,
                                                DIM, NG / 16, 8, 1);
    k_copy<<<g1d((long)NG * 128), B, 0, stream>>>(outp + NG, G, (long)NG * 128);

    // ---- target branch ----
    k_col_stats<<<TD, B, 0, stream>>>(target, TS, NG, TD);
    k_bn_apply<<<g1d((long)NG * TD), B, 0, stream>>>(target, TN, TS, NG, TD, g1, be1);
    k_gemm<<<(128 * 11 + 7) / 8, B, 0, stream>>>(TN, TD, WP31, 176, BP31, T1, 176,
                                                 TD, NG / 16, 11, 0);
    k_gemm<<<(128 * 8 + 7) / 8, B, 0, stream>>>(T1, 176, WP32, 128, b32, T2, 128,
                                                176, NG / 16, 8, 0);
    k_softmax128<<<(NG * 32) / B, B, 0, stream>>>(T2, NG);

    // ---- fusion head ----
    k_build_xc<<<g1d((long)NG * 256), B, 0, stream>>>(XC, G, T2);
    k_gemm<<<(128 * 8 + 7) / 8, B, 0, stream>>>(XC, 256, W4, 128, b4, XC2, 128,
                                                256, NG / 16, 8, 1);
    k_head<<<(NG * 32) / B, B, 0, stream>>>(XC2, W5, b5, outp, NG);
}